// ALIGNN_JP_3332894621747
// MI455X (gfx1250) — compile-verified
//
#include <hip/hip_runtime.h>
#include <hip/hip_bf16.h>
#include <math.h>
#include <stdint.h>

#define N_ATOMS 5000
#define N_EDGES 40000
#define N_TRIP  250000
#define NHEAD   4
#define HIDDIM  64
#define OUTDIM  64
#define NTYPES  108
#define KDIM    (NTYPES * NHEAD)   /* 432 */
#define MPAD    5008               /* 313 tiles of 16 */
#define EPSV    0.001f
#define BCOUNT  (NTYPES * OUTDIM * NHEAD)  /* 27648 floats = 110,592 B */

typedef __attribute__((ext_vector_type(2))) float v2f;
typedef __attribute__((ext_vector_type(8))) float v8f;

// ---------------------------------------------------------------------------
// Stage 1: per-triplet. s[h] = weight[h]*spatial[h] atomically accumulated per
// source line-graph edge. Key tables (108KB) stay hot in WGP$.
// ---------------------------------------------------------------------------
__global__ void alignn_triplet_kernel(
    const float* __restrict__ r, const float* __restrict__ dnr,
    const float* __restrict__ key_emb,
    const float* __restrict__ Ap, const float* __restrict__ Bp,
    const float* __restrict__ Cp, const float* __restrict__ Dp,
    const int* __restrict__ src_idx, const int* __restrict__ dst_idx,
    const int* __restrict__ lg_src, const int* __restrict__ lg_dst,
    const int* __restrict__ atomic_number,
    float* __restrict__ acc, float* __restrict__ cnt_lg)
{
    int t = blockIdx.x * blockDim.x + threadIdx.x;
    if (t >= N_TRIP) return;

    int ea = lg_src[t];
    int ec = lg_dst[t];
    int za = atomic_number[src_idx[ea]];
    int zc = atomic_number[dst_idx[ec]];

    // weight[h] = sum_hid key_emb[za][hid*4+h] * key_emb[zc][hid*4+h]
    const float4* ka = (const float4*)(key_emb + (size_t)za * (HIDDIM * NHEAD));
    const float4* kc = (const float4*)(key_emb + (size_t)zc * (HIDDIM * NHEAD));
    float wx = 0.f, wy = 0.f, wz = 0.f, ww = 0.f;
#pragma unroll 8
    for (int i = 0; i < HIDDIM; ++i) {
        float4 x = ka[i];
        float4 y = kc[i];
        wx = fmaf(x.x, y.x, wx);
        wy = fmaf(x.y, y.y, wy);
        wz = fmaf(x.z, y.z, wz);
        ww = fmaf(x.w, y.w, ww);
    }

    // angular term
    float r1x = -r[ea * 3 + 0], r1y = -r[ea * 3 + 1], r1z = -r[ea * 3 + 2];
    float r2x =  r[ec * 3 + 0], r2y =  r[ec * 3 + 1], r2z =  r[ec * 3 + 2];
    float dot = r1x * r2x + r1y * r2y + r1z * r2z;
    float n1  = sqrtf(r1x * r1x + r1y * r1y + r1z * r1z);
    float n2  = sqrtf(r2x * r2x + r2y * r2y + r2z * r2z);
    float cosv = dot / (n1 * n2);
    cosv = fminf(fmaxf(cosv, -1.0f), 1.0f);
    cosv = fminf(fmaxf(cosv, -EPSV), EPSV);
    float theta = acosf(cosv);

    float dn  = dnr[t];
    float dn2 = dn * dn;

    float w4[4] = { wx, wy, wz, ww };
    float s4[4];
#pragma unroll
    for (int h = 0; h < NHEAD; ++h) {
        float Bm  = fmodf(Bp[h], 3.14159265358979323846f);
        float ang = (cosf(Ap[h] * theta + Bm) + 1.0f) * 0.5f;
        ang = powf(ang, Cp[h]);
        float rad = expf(-Dp[h] * dn2);
        s4[h] = w4[h] * ang * rad;
    }

    float* ap = acc + (size_t)ea * 4;
    atomicAdd(&ap[0], s4[0]);
    atomicAdd(&ap[1], s4[1]);
    atomicAdd(&ap[2], s4[2]);
    atomicAdd(&ap[3], s4[3]);
    atomicAdd(&cnt_lg[ea], 1.0f);
}

// ---------------------------------------------------------------------------
// Stage 2: per-edge. g[e,h] = acc/max(cnt,1); scatter into type-bucketed P.
// ---------------------------------------------------------------------------
__global__ void alignn_edge_kernel(
    const int* __restrict__ src_idx, const int* __restrict__ dst_idx,
    const int* __restrict__ atomic_number,
    const float* __restrict__ acc, const float* __restrict__ cnt_lg,
    float* __restrict__ Pmat, float* __restrict__ cnt_atom)
{
    int e = blockIdx.x * blockDim.x + threadIdx.x;
    if (e >= N_EDGES) return;

    float inv = 1.0f / fmaxf(cnt_lg[e], 1.0f);
    int n = src_idx[e];
    int z = atomic_number[dst_idx[e]];
    const float* ap = acc + (size_t)e * 4;
    float* pr = Pmat + (size_t)n * KDIM + z * 4;
    atomicAdd(&pr[0], ap[0] * inv);
    atomicAdd(&pr[1], ap[1] * inv);
    atomicAdd(&pr[2], ap[2] * inv);
    atomicAdd(&pr[3], ap[3] * inv);
    atomicAdd(&cnt_atom[n], 1.0f);
}

// ---------------------------------------------------------------------------
// Stage 3: out[n,o] = (1/cnt_n) * sum_k P[n,k] * Vt[k,o]   via FP32 WMMA.
// Vt[z*4+h][o] = value_emb[z*256 + o*4 + h]  — copied verbatim into LDS once
// per workgroup via CDNA5 async load-to-LDS (ASYNCcnt path), then the WMMA
// B-fragments come from ds_load_b64 with immediate offsets. Only the A
// fragment (streaming P rows) stays on the VMEM path.
// ---------------------------------------------------------------------------
__global__ void alignn_gemm_wmma_kernel(
    const float* __restrict__ Pmat, const float* __restrict__ value_emb,
    const float* __restrict__ cnt_atom, float* __restrict__ out)
{
    __shared__ float Bs[BCOUNT];  // 110,592 bytes < 320KB WGP LDS

    // --- async stage value_emb -> LDS (all threads participate) ---
    {
        const int nvec = BCOUNT / 4;  // 6912 float4 transfers
        for (int i = threadIdx.x; i < nvec; i += blockDim.x) {
            uint32_t lds_off = (uint32_t)(uintptr_t)(&Bs[i * 4]);
            const float* gsrc = value_emb + (size_t)i * 4;
            asm volatile("global_load_async_to_lds_b128 %0, %1, off"
                         :: "v"(lds_off), "v"(gsrc)
                         : "memory");
        }
        asm volatile("s_wait_asynccnt 0x0" ::: "memory");
        __syncthreads();
    }

    const int lane  = threadIdx.x & 31;
    const int wave  = threadIdx.x >> 5;
    const int tile  = blockIdx.x * (blockDim.x >> 5) + wave;
    const int mtiles = MPAD / 16; // 313
    if (tile >= mtiles) return;   // wave-uniform: EXEC stays all-ones for WMMA

    const int m0    = tile * 16;
    const int mrow  = m0 + (lane & 15);        // A-frag row for this lane
    const int koff  = (lane >> 4) << 1;        // 0 for lanes 0-15, 2 for 16-31
    const int col_l = lane & 15;

    v8f acc0 = {}, acc1 = {}, acc2 = {}, acc3 = {};

    const float* prow = Pmat + (size_t)mrow * KDIM;
    const float* bbase = Bs + col_l * 4 + koff;
#pragma unroll 4
    for (int k = 0; k < KDIM; k += 4) {
        const int z = k >> 2;
        v2f afrag = *(const v2f*)(prow + k + koff);
        const float* vrow = bbase + z * (OUTDIM * NHEAD);
        v2f b0 = *(const v2f*)(vrow +   0);
        v2f b1 = *(const v2f*)(vrow +  64);
        v2f b2 = *(const v2f*)(vrow + 128);
        v2f b3 = *(const v2f*)(vrow + 192);
        acc0 = __builtin_amdgcn_wmma_f32_16x16x4_f32(false, afrag, false, b0,
                                                     (short)0, acc0, false, false);
        acc1 = __builtin_amdgcn_wmma_f32_16x16x4_f32(false, afrag, false, b1,
                                                     (short)0, acc1, false, false);
        acc2 = __builtin_amdgcn_wmma_f32_16x16x4_f32(false, afrag, false, b2,
                                                     (short)0, acc2, false, false);
        acc3 = __builtin_amdgcn_wmma_f32_16x16x4_f32(false, afrag, false, b3,
                                                     (short)0, acc3, false, false);
    }

    // D layout: VGPR v, lanes 0-15 -> row m0+v, lanes 16-31 -> row m0+8+v
    const int rbase = m0 + ((lane >> 4) << 3);
#pragma unroll
    for (int v = 0; v < 8; ++v) {
        int rr = rbase + v;
        if (rr < N_ATOMS) {
            float s = 1.0f / fmaxf(cnt_atom[rr], 1.0f);
            float* orow = out + (size_t)rr * OUTDIM;
            orow[ 0 + col_l] = acc0[v] * s;
            orow[16 + col_l] = acc1[v] * s;
            orow[32 + col_l] = acc2[v] * s;
            orow[48 + col_l] = acc3[v] * s;
        }
    }
}

extern "C" void kernel_launch(void* const* d_in, const int* in_sizes, int n_in,
                              void* d_out, int out_size, void* d_ws, size_t ws_size,
                              hipStream_t stream) {
    const float* r             = (const float*)d_in[0];
    const float* dnr           = (const float*)d_in[1];
    const float* key_emb       = (const float*)d_in[2];
    const float* value_emb     = (const float*)d_in[3];
    const float* Ap            = (const float*)d_in[4];
    const float* Bp            = (const float*)d_in[5];
    const float* Cp            = (const float*)d_in[6];
    const float* Dp            = (const float*)d_in[7];
    const int*   src_idx       = (const int*)d_in[8];
    const int*   dst_idx       = (const int*)d_in[9];
    const int*   lg_src        = (const int*)d_in[10];
    const int*   lg_dst        = (const int*)d_in[11];
    const int*   atomic_number = (const int*)d_in[12];
    float*       out           = (float*)d_out;

    // Workspace layout (floats): acc[E*4] | cnt_lg[E] | P[MPAD*KDIM] | cnt_atom[N]
    float* ws       = (float*)d_ws;
    float* acc      = ws;
    float* cnt_lg   = acc + (size_t)N_EDGES * 4;
    float* Pmat     = cnt_lg + N_EDGES;
    float* cnt_atom = Pmat + (size_t)MPAD * KDIM;
    size_t zero_f   = (size_t)N_EDGES * 4 + N_EDGES + (size_t)MPAD * KDIM + N_ATOMS;

    hipMemsetAsync(d_ws, 0, zero_f * sizeof(float), stream);

    alignn_triplet_kernel<<<(N_TRIP + 255) / 256, 256, 0, stream>>>(
        r, dnr, key_emb, Ap, Bp, Cp, Dp,
        src_idx, dst_idx, lg_src, lg_dst, atomic_number, acc, cnt_lg);

    alignn_edge_kernel<<<(N_EDGES + 255) / 256, 256, 0, stream>>>(
        src_idx, dst_idx, atomic_number, acc, cnt_lg, Pmat, cnt_atom);

    const int mtiles = MPAD / 16;            // 313 waves
    const int wpb    = 8;                    // 256 threads = 8 waves per block
    alignn_gemm_wmma_kernel<<<(mtiles + wpb - 1) / wpb, wpb * 32, 0, stream>>>(
        Pmat, value_emb, cnt_atom, out);
}